// GCONVDIFF_WORDS_60224031425326
// MI455X (gfx1250) — compile-verified
//
#include <hip/hip_runtime.h>
#include <math.h>

// CDNA5 / gfx1250. wave32. f32 WMMA 16x16x4 for the dual GEMM.

typedef __attribute__((ext_vector_type(2))) float v2f;
typedef __attribute__((ext_vector_type(8))) float v8f;

#define DFEAT 128
#define LDS_PITCH 132   // 128 + 4 pad -> conflict-free A-frag reads (bank = (4r+c)%64)

// ---------------------------------------------------------------- zero agg
__global__ void gc_zero_kernel(float* __restrict__ p, int n4) {
    int i = blockIdx.x * blockDim.x + threadIdx.x;
    if (i < n4) ((float4*)p)[i] = make_float4(0.f, 0.f, 0.f, 0.f);
}

// ------------------------------------------------- segment-sum over edges
// lane = 4-float chunk of one edge -> one coalesced 512B row per edge read,
// f32 atomic adds into agg (L2-resident, 51MB << 192MB L2).
__global__ void gc_scatter_kernel(const float* __restrict__ x,
                                  const int* __restrict__ eidx,
                                  float* __restrict__ agg,
                                  int E) {
    int gid = blockIdx.x * blockDim.x + threadIdx.x;
    int e = gid >> 5;
    if (e >= E) return;
    int c = (gid & 31) << 2;
    int s = eidx[e];        // src row
    int d = eidx[E + e];    // dst row
    const float4 v = *(const float4*)(x + (size_t)s * DFEAT + c);
    float* ap = agg + (size_t)d * DFEAT + c;
    atomicAdd(ap + 0, v.x);
    atomicAdd(ap + 1, v.y);
    atomicAdd(ap + 2, v.z);
    atomicAdd(ap + 3, v.w);
}

// ------------------------------------------- fused dual GEMM + bias + relu
// out[n,h] = relu( sum_d agg[n,d]*Wrel[h,d] + brel[h] + sum_d x[n,d]*Wroot[h,d] )
// Block: 256 threads = 8 waves. Block owns 16 node rows; wave w owns output
// columns [16w, 16w+16). K-loop: 32 steps of V_WMMA_F32_16X16X4_F32, two per
// step (rel + root) into one f32 accumulator. Writes IN PLACE over agg
// (rows staged to LDS first; rows are block-exclusive).
__global__ void __launch_bounds__(256) gc_gemm_kernel(
    float* __restrict__ agg,            // in: aggregate, out: new features
    const float* __restrict__ xin,      // root-path input
    const float* __restrict__ Wrel,     // [D][D] row-major (h,d)
    const float* __restrict__ brel,     // [D]
    const float* __restrict__ Wroot,    // [D][D]
    int N)
{
    __shared__ float shA[16 * LDS_PITCH];
    __shared__ float shX[16 * LDS_PITCH];

    const int tid = threadIdx.x;
    const int tileBase = blockIdx.x * 16;

    // cooperative stage: 16 rows x 128 cols of agg and xin (two float4 each)
    #pragma unroll
    for (int half = 0; half < 2; ++half) {
        int flat = (tid + half * 256) * 4;       // 0..2044
        int r = flat >> 7;
        int c = flat & 127;
        int row = tileBase + r;
        float4 va, vx;
        if (row < N) {
            va = *(const float4*)(agg + (size_t)row * DFEAT + c);
            vx = *(const float4*)(xin + (size_t)row * DFEAT + c);
        } else {
            va = make_float4(0.f, 0.f, 0.f, 0.f);
            vx = va;
        }
        *(float4*)(shA + r * LDS_PITCH + c) = va;
        *(float4*)(shX + r * LDS_PITCH + c) = vx;
    }
    __syncthreads();

    const int lane = tid & 31;
    const int wave = tid >> 5;
    const int colBase = wave * 16;
    const int mn   = lane & 15;            // A: row M in tile / B: col N in tile
    const int koff = (lane >> 4) << 1;     // lanes 0-15 -> K{0,1}, 16-31 -> K{2,3}

    const float* wrelRow  = Wrel  + (size_t)(colBase + mn) * DFEAT + koff;
    const float* wrootRow = Wroot + (size_t)(colBase + mn) * DFEAT + koff;
    const float* shArow = shA + mn * LDS_PITCH + koff;
    const float* shXrow = shX + mn * LDS_PITCH + koff;

    v8f acc = {};
    #pragma unroll
    for (int kb = 0; kb < DFEAT; kb += 4) {
        v2f aA = *(const v2f*)(shArow + kb);
        v2f aX = *(const v2f*)(shXrow + kb);
        v2f bR = *(const v2f*)(wrelRow + kb);
        v2f bT = *(const v2f*)(wrootRow + kb);
        acc = __builtin_amdgcn_wmma_f32_16x16x4_f32(
                  false, aA, false, bR, (short)0, acc, false, false);
        acc = __builtin_amdgcn_wmma_f32_16x16x4_f32(
                  false, aX, false, bT, (short)0, acc, false, false);
    }

    // epilogue: +bias, relu, in-place store. C/D layout: vgpr v, lane l ->
    // M = (l>=16 ? 8:0)+v, Ncol = l%16
    const float bias = brel[colBase + mn];
    const int rowHalf = tileBase + ((lane >> 4) << 3);
    #pragma unroll
    for (int v = 0; v < 8; ++v) {
        int row = rowHalf + v;
        float val = fmaxf(acc[v] + bias, 0.f);
        if (row < N) agg[(size_t)row * DFEAT + colBase + mn] = val;
    }
}

// ------------------------------------------- pairwise sqrt-diff + sigmoid
// one 128-thread block per pair p: x1=x[2p], x2=x[2p+1]
__global__ void __launch_bounds__(128) gc_final_kernel(
    const float* __restrict__ xf,
    const float* __restrict__ Wlin,    // [D]
    const float* __restrict__ blin,    // [1]
    float* __restrict__ outbuf,        // d_out: sig | out | x1 | x2
    int P)
{
    __shared__ float red[128];
    int p = blockIdx.x;
    int h = threadIdx.x;
    float a = xf[(size_t)(2 * p) * DFEAT + h];
    float b = xf[(size_t)(2 * p + 1) * DFEAT + h];
    float dd = a - b;
    float o = sqrtf(dd * dd + 0.001f);

    float* sig  = outbuf;
    float* outm = outbuf + P;
    float* x1m  = outbuf + P + (size_t)P * DFEAT;
    float* x2m  = outbuf + P + (size_t)2 * P * DFEAT;
    outm[(size_t)p * DFEAT + h] = o;
    x1m[(size_t)p * DFEAT + h] = a;
    x2m[(size_t)p * DFEAT + h] = b;

    red[h] = o * Wlin[h];
    __syncthreads();
    #pragma unroll
    for (int s = 64; s > 0; s >>= 1) {
        if (h < s) red[h] += red[h + s];
        __syncthreads();
    }
    if (h == 0) {
        float z = red[0] + blin[0];
        sig[p] = 1.f / (1.f + expf(-z));
    }
}

extern "C" void kernel_launch(void* const* d_in, const int* in_sizes, int n_in,
                              void* d_out, int out_size, void* d_ws, size_t ws_size,
                              hipStream_t stream) {
    const float* x     = (const float*)d_in[0];
    const int*   eidx  = (const int*)d_in[1];   // int64 in ref downcasts to i32 (jax default)
    // d_in[2] = batch (unused: single graph)
    const float* Wrel  = (const float*)d_in[3];
    const float* brel  = (const float*)d_in[4];
    const float* Wroot = (const float*)d_in[5];
    const float* Wlin  = (const float*)d_in[6];
    const float* blin  = (const float*)d_in[7];

    const int N  = in_sizes[0] / DFEAT;
    const int E  = in_sizes[1] / 2;
    const int NC = in_sizes[4] / DFEAT;

    float* bufA = (float*)d_ws;
    float* bufB = bufA + (size_t)N * DFEAT;

    const int n4         = N * DFEAT / 4;
    const int zeroBlocks = (n4 + 255) / 256;
    const int scatBlocks = (int)(((long long)E * 32 + 255) / 256);
    const int gemmBlocks = (N + 15) / 16;

    const float* cur = x;
    float* bufs[2] = {bufA, bufB};
    for (int i = 0; i < NC; ++i) {
        float* agg = bufs[i & 1];
        gc_zero_kernel<<<zeroBlocks, 256, 0, stream>>>(agg, n4);
        gc_scatter_kernel<<<scatBlocks, 256, 0, stream>>>(cur, eidx, agg, E);
        gc_gemm_kernel<<<gemmBlocks, 256, 0, stream>>>(
            agg, cur,
            Wrel  + (size_t)i * DFEAT * DFEAT,
            brel  + (size_t)i * DFEAT,
            Wroot + (size_t)i * DFEAT * DFEAT,
            N);
        cur = agg;
    }

    const int P = N / 2;
    gc_final_kernel<<<P, 128, 0, stream>>>(cur, Wlin, blin, (float*)d_out, P);
}